// PrevSamplesEmbedder_1417339208140
// MI455X (gfx1250) — compile-verified
//
#include <hip/hip_runtime.h>

typedef float v2f __attribute__((ext_vector_type(2)));
typedef float v8f __attribute__((ext_vector_type(8)));

#define NB   4096
#define DD   256
#define NKEY 512
#define KDIM 16
#define NQ   4
#define SED  16
// q projection output = NQ*KDIM = 64

__global__ __launch_bounds__(256) void psemb_kernel(
    const float* __restrict__ x,      // [B, 256]
    const float* __restrict__ keys,   // [B, 512, 16]
    const float* __restrict__ values, // [B, 512, 16]
    const float* __restrict__ Wq,     // [A, 256, 64]
    const float* __restrict__ bq,     // [A, 64]
    const int*   __restrict__ addr,   // [B]
    float*       __restrict__ out)    // [B, 64]
{
    __shared__ float xs[256];        // x[b]
    __shared__ float qs[256];        // q padded to 16x16, rows 4..15 zero
    __shared__ float sc[NQ * NKEY];  // scores -> attn, 4x512
    __shared__ float red[256];       // reduction scratch
    __shared__ float part[8 * 64];   // per-wave AV partials

    const int b    = blockIdx.x;
    const int tid  = threadIdx.x;
    const int lane = tid & 31;       // wave32
    const int wave = tid >> 5;       // 8 waves
    const int a    = addr[b];

    // ---- Stage 1: routed query projection q = x[b] @ Wq[a] + bq[a] ----
    xs[tid] = x[(size_t)b * DD + tid];
    qs[tid] = 0.0f;                  // zero-pad A-matrix rows 4..15
    __syncthreads();
    {
        const int k  = tid & 63;     // output column (coalesced across lanes)
        const int ch = tid >> 6;     // quarter of D
        const float* wp = Wq + (size_t)a * (DD * 64) + (size_t)(ch * 64) * 64 + k;
        float p = 0.0f;
        #pragma unroll 8
        for (int d = 0; d < 64; ++d)
            p = fmaf(xs[ch * 64 + d], wp[(size_t)d * 64], p);
        red[tid] = p;
    }
    __syncthreads();
    if (tid < 64)
        qs[tid] = red[tid] + red[tid + 64] + red[tid + 128] + red[tid + 192]
                + bq[(size_t)a * 64 + tid];
    __syncthreads();

    // ---- Stage 2: scores[4,512] = q @ keys^T via chained WMMA f32 16x16x4 ----
    {
        const int m  = lane & 15;    // A row (query) == B col (key-in-tile)
        const int kh = lane >> 4;    // lane half selects K 0/1 vs 2/3
        const float* arow = qs + m * 16 + 2 * kh;
        for (int t = wave; t < NKEY / 16; t += 8) {
            const int base = t * 16;
            const float* krow = keys + ((size_t)b * NKEY + base + m) * KDIM + 2 * kh;
            v8f c = {};
            #pragma unroll
            for (int kk = 0; kk < 4; ++kk) {
                v2f af = *(const v2f*)(arow + 4 * kk);
                v2f bf = *(const v2f*)(krow + 4 * kk);
                c = __builtin_amdgcn_wmma_f32_16x16x4_f32(
                        false, af, false, bf, (short)0, c, false, false);
            }
            if (lane < 16) {         // D rows 0..3 live in VGPR 0..3, lanes 0..15
                #pragma unroll
                for (int r = 0; r < NQ; ++r)
                    sc[r * NKEY + base + lane] = c[r];
            }
        }
    }
    __syncthreads();

    // ---- Stage 3: softmax over 512 keys for each of 4 queries ----
    {
        const int q = tid >> 6;      // 64 threads per query
        const int j = tid & 63;
        float mx = -3.4e38f;
        for (int n = j; n < NKEY; n += 64) mx = fmaxf(mx, sc[q * NKEY + n]);
        red[tid] = mx;
        __syncthreads();
        for (int off = 32; off >= 1; off >>= 1) {
            if (j < off) red[tid] = fmaxf(red[tid], red[tid + off]);
            __syncthreads();
        }
        const float mxv = red[q * 64];
        __syncthreads();
        float s = 0.0f;
        for (int n = j; n < NKEY; n += 64) {
            const float e = __expf(sc[q * NKEY + n] - mxv);
            sc[q * NKEY + n] = e;
            s += e;
        }
        red[tid] = s;
        __syncthreads();
        for (int off = 32; off >= 1; off >>= 1) {
            if (j < off) red[tid] += red[tid + off];
            __syncthreads();
        }
        const float inv = 1.0f / red[q * 64];
        __syncthreads();
        for (int n = j; n < NKEY; n += 64) sc[q * NKEY + n] *= inv;
    }
    __syncthreads();

    // ---- Stage 4: out = attn @ values via chained WMMA, 64 keys per wave ----
    {
        const int m   = lane & 15;   // A row (query) == B col (sed dim)
        const int kh  = lane >> 4;
        const int row = (m < NQ) ? m : 0;
        const int kb0 = wave * 64;
        v8f c = {};
        for (int kt = 0; kt < 16; ++kt) {
            const int kb = kb0 + kt * 4;
            v2f af = *(const v2f*)(sc + row * NKEY + kb + 2 * kh);
            af[0] = (m < NQ) ? af[0] : 0.0f;   // selects, EXEC unchanged
            af[1] = (m < NQ) ? af[1] : 0.0f;
            const float* vrow = values + ((size_t)b * NKEY + kb + 2 * kh) * SED + m;
            v2f bf = { vrow[0], vrow[SED] };
            c = __builtin_amdgcn_wmma_f32_16x16x4_f32(
                    false, af, false, bf, (short)0, c, false, false);
        }
        if (lane < 16) {
            #pragma unroll
            for (int r = 0; r < NQ; ++r)
                part[wave * 64 + r * 16 + lane] = c[r];
        }
    }
    __syncthreads();

    if (tid < 64) {
        float s = 0.0f;
        #pragma unroll
        for (int w = 0; w < 8; ++w) s += part[w * 64 + tid];
        out[(size_t)b * 64 + tid] = s;
    }
}

extern "C" void kernel_launch(void* const* d_in, const int* in_sizes, int n_in,
                              void* d_out, int out_size, void* d_ws, size_t ws_size,
                              hipStream_t stream) {
    (void)in_sizes; (void)n_in; (void)d_ws; (void)ws_size; (void)out_size;
    const float* x      = (const float*)d_in[0];
    const float* keys   = (const float*)d_in[1];
    const float* values = (const float*)d_in[2];
    const float* Wq     = (const float*)d_in[3];
    const float* bq     = (const float*)d_in[4];
    const int*   addr   = (const int*)d_in[5];
    float* out = (float*)d_out;
    psemb_kernel<<<dim3(NB), dim3(256), 0, stream>>>(x, keys, values, Wq, bq, addr, out);
}